// AdaptiveResonateAndFireNeuron_8108898255618
// MI455X (gfx1250) — compile-verified
//
#include <hip/hip_runtime.h>

// Adaptive resonate-and-fire neuron, MI455X (gfx1250, wave32).
// Phase 1: proj = x @ W^T + b  via V_WMMA_F32_16X16X4_F32 (fp32-exact vs reference).
// Phase 2: 1024-step leaky-integrate scan, one wave per batch, neurons in registers,
//          inhibition reduction via 5x shfl_xor (no LDS / barriers on the critical chain).

typedef float v2f __attribute__((ext_vector_type(2)));
typedef float v8f __attribute__((ext_vector_type(8)));

#define B_DIM 32
#define S_DIM 1024
#define I_DIM 512
#define O_DIM 512

#define DECAY        0.9f
#define INHIB        0.1f
#define BASE_THRESH  1.0f
#define THRESH_DECAY 0.9f
#define THRESH_INC   0.1f

// ---------------------------------------------------------------------------
// GEMM: proj[m][n] = sum_k X[m][k] * W[n][k] + bias[n]
// M = B*S = 32768 (m = b*S + s), N = O = 512, K = I = 512.
// Block = 256 threads = 8 waves (2 in M x 4 in N). Wave tile = 64x64 = 4x4 WMMAs.
// Workgroup tile = 128 (M) x 256 (N); grid = (N/256, M/128) = (2, 256).
//
// f32 WMMA fragment mapping (wave32):
//   A 16x4 :  lane l<16 holds A[m=l][k=0..1], lane l>=16 holds A[m=l-16][k=2..3]
//   B 4x16 :  lane l<16 holds B[k=0..1][n=l], lane l>=16 holds B[k=2..3][n=l-16]
// With W stored [n][k], both fragments are the same float2 gather from K-major rows.
//   C 16x16: VGPR r -> row (r + 8*(lane>=16)), col (lane&15).
// ---------------------------------------------------------------------------
__global__ __launch_bounds__(256)
void arf_proj_wmma_f32(const float* __restrict__ X,
                       const float* __restrict__ W,
                       const float* __restrict__ bias,
                       float* __restrict__ proj)
{
    const int lane = threadIdx.x & 31;
    const int wave = threadIdx.x >> 5;
    const int wm   = wave >> 2;          // 0..1
    const int wn   = wave & 3;           // 0..3

    const int m_base = blockIdx.y * 128 + wm * 64;
    const int n_base = blockIdx.x * 256 + wn * 64;

    const int ml = lane & 15;            // row (A) / col (B) within 16-wide tile
    const int kb = (lane >> 4) << 1;     // K sub-offset: 0 for lanes 0-15, 2 for 16-31
    const int rhi = (lane >> 4) << 3;    // +8 row offset for C in upper half-wave

    v8f acc[4][4] = {};

    const float* arow[4];
    const float* brow[4];
#pragma unroll
    for (int i = 0; i < 4; ++i)
        arow[i] = X + (size_t)(m_base + i * 16 + ml) * I_DIM + kb;
#pragma unroll
    for (int j = 0; j < 4; ++j)
        brow[j] = W + (size_t)(n_base + j * 16 + ml) * I_DIM + kb;

    for (int k = 0; k < I_DIM; k += 4) {
        if ((k & 127) == 0) {
#pragma unroll
            for (int i = 0; i < 4; ++i)
                __builtin_prefetch(arow[i] + k + 128, 0, 1);
        }
        v2f a[4], b[4];
#pragma unroll
        for (int i = 0; i < 4; ++i) a[i] = *(const v2f*)(arow[i] + k);
#pragma unroll
        for (int j = 0; j < 4; ++j) b[j] = *(const v2f*)(brow[j] + k);
#pragma unroll
        for (int i = 0; i < 4; ++i)
#pragma unroll
            for (int j = 0; j < 4; ++j)
                acc[i][j] = __builtin_amdgcn_wmma_f32_16x16x4_f32(
                    /*neg_a=*/false, a[i], /*neg_b=*/false, b[j],
                    /*c_mod=*/(short)0, acc[i][j],
                    /*reuse_a=*/false, /*reuse_b=*/false);
    }

    // Epilogue: bias add + store (proj is [B][S][O] row-major, stride O_DIM).
#pragma unroll
    for (int j = 0; j < 4; ++j) {
        const int n  = n_base + j * 16 + ml;
        const float bb = bias[n];
#pragma unroll
        for (int i = 0; i < 4; ++i) {
            const int mrow = m_base + i * 16 + rhi;
#pragma unroll
            for (int r = 0; r < 8; ++r)
                proj[(size_t)(mrow + r) * O_DIM + n] = acc[i][j][r] + bb;
        }
    }
}

// ---------------------------------------------------------------------------
// Scan: one wave (32 lanes) per batch; each lane owns 16 consecutive neurons.
// Per step:   inhib = INHIB * sum_o mem_prev   (local 16-add + 5 shfl_xor hops)
//             cur   = proj[s] - inhib
//             mem   = DECAY*mem + cur;  spike = (mem >= thr)
//             mem  -= spike*thr;        thr = THRESH_DECAY*thr + THRESH_INC*spike
// proj/out streamed as 4x float4 per lane per step (fully coalesced 2KiB/wave).
// ---------------------------------------------------------------------------
__global__ __launch_bounds__(32)
void arf_scan(const float* __restrict__ proj, float* __restrict__ out)
{
    const int b    = blockIdx.x;
    const int lane = threadIdx.x;

    const float* p = proj + (size_t)b * S_DIM * O_DIM + lane * 16;
    float*       o = out  + (size_t)b * S_DIM * O_DIM + lane * 16;

    float mem[16], thr[16];
#pragma unroll
    for (int j = 0; j < 16; ++j) { mem[j] = 0.0f; thr[j] = BASE_THRESH; }

    for (int s = 0; s < S_DIM; ++s) {
        const float4* prow = (const float4*)(p + (size_t)s * O_DIM);
        float4 c[4];
#pragma unroll
        for (int q = 0; q < 4; ++q) c[q] = prow[q];

        // Inhibition uses mem from BEFORE this step's update.
        float part = 0.0f;
#pragma unroll
        for (int j = 0; j < 16; ++j) part += mem[j];
#pragma unroll
        for (int mask = 16; mask >= 1; mask >>= 1)
            part += __shfl_xor(part, mask, 32);
        const float inhib = INHIB * part;

        float cur[16] = { c[0].x, c[0].y, c[0].z, c[0].w,
                          c[1].x, c[1].y, c[1].z, c[1].w,
                          c[2].x, c[2].y, c[2].z, c[2].w,
                          c[3].x, c[3].y, c[3].z, c[3].w };
        float spk[16];
#pragma unroll
        for (int j = 0; j < 16; ++j) {
            float m     = DECAY * mem[j] + (cur[j] - inhib);
            float spike = (m >= thr[j]) ? 1.0f : 0.0f;
            m          -= spike * thr[j];
            thr[j]      = THRESH_DECAY * thr[j] + THRESH_INC * spike;
            mem[j]      = m;
            spk[j]      = spike;
        }

        float4* orow = (float4*)(o + (size_t)s * O_DIM);
#pragma unroll
        for (int q = 0; q < 4; ++q)
            orow[q] = make_float4(spk[4*q+0], spk[4*q+1], spk[4*q+2], spk[4*q+3]);
    }
}

// ---------------------------------------------------------------------------
extern "C" void kernel_launch(void* const* d_in, const int* in_sizes, int n_in,
                              void* d_out, int out_size, void* d_ws, size_t ws_size,
                              hipStream_t stream)
{
    (void)in_sizes; (void)n_in; (void)out_size; (void)ws_size;

    const float* x    = (const float*)d_in[0];   // [B, S, I]
    const float* W    = (const float*)d_in[1];   // [O, I]
    const float* bias = (const float*)d_in[2];   // [O]
    float*       proj = (float*)d_ws;            // [B, S, O] = 64 MiB scratch
    float*       outp = (float*)d_out;           // [B, S, O]

    // GEMM: grid (N/256, M/128) = (2, 256), 256 threads (8 waves).
    dim3 gemm_grid(O_DIM / 256, (B_DIM * S_DIM) / 128);
    arf_proj_wmma_f32<<<gemm_grid, 256, 0, stream>>>(x, W, bias, proj);

    // Scan: one wave per batch.
    arf_scan<<<B_DIM, 32, 0, stream>>>(proj, outp);
}